// SpatialAARMAdapter_7215545057651
// MI455X (gfx1250) — compile-verified
//
#include <hip/hip_runtime.h>
#include <math.h>

typedef float v2f __attribute__((ext_vector_type(2)));
typedef float v4f __attribute__((ext_vector_type(4)));
typedef float v8f __attribute__((ext_vector_type(8)));

#define CCH   64          // channels (C1 == C2)
#define NPIX  16384       // H*W
#define NB    32          // batch
#define WGPB  8           // workgroups per batch in gram pass
#define CHUNK 64          // pixels staged in LDS per iteration
#define LROW  68          // padded LDS row stride (floats): 16B-aligned rows,
                          // conflict-free frag reads (4*(lane%16) mod 64 distinct)

// ---------------------------------------------------------------------------
// Kernel 1: per-batch channel Gram G = X·X^T and row sums s = X·1.
// Each WG covers 2048 pixels; 8 waves each own 2 k-steps (of 4 px) per 64-px
// chunk; every wave holds the full 4x4 grid of 16x16 f32 accumulator tiles.
// Staging uses b128 global loads + b128 LDS stores.
// ---------------------------------------------------------------------------
__global__ __launch_bounds__(256) void gram_kernel(const float* __restrict__ x,
                                                   float* __restrict__ gpart,
                                                   float* __restrict__ spart) {
  __shared__ __align__(16) float xs[CCH * LROW];  // 64 x 68 floats
  __shared__ float gs[CCH * CCH];                 // 64 x 64 reduction buffer

  const int b    = blockIdx.x / WGPB;
  const int wg   = blockIdx.x % WGPB;
  const int tid  = threadIdx.x;
  const int lane = tid & 31;
  const int wave = tid >> 5;
  const int lh   = lane >> 4;        // lane half (K pair select)
  const int l16  = lane & 15;        // M / N index within tile

  const float* xb = x + (size_t)b * CCH * NPIX;
  const int p0 = wg * (NPIX / WGPB);

  const v8f zacc = {0.f, 0.f, 0.f, 0.f, 0.f, 0.f, 0.f, 0.f};
  v8f acc[4][4];
  for (int i = 0; i < 4; ++i)
    for (int j = 0; j < 4; ++j) acc[i][j] = zacc;

  float srow = 0.f;  // thread t<64 accumulates row-sum of channel t

  for (int ch = 0; ch < (NPIX / WGPB) / CHUNK; ++ch) {  // 32 chunks
    const int pbase = p0 + ch * CHUNK;
    // Coalesced b128 stage of X[0:64][pbase:pbase+64] into LDS.
    // 1024 float4s total; 4 per thread. row = e>>4, col4 = e&15.
    for (int i = 0; i < 4; ++i) {
      int e = tid + 256 * i;
      int r = e >> 4, c4 = e & 15;
      v4f v = ((const v4f*)(xb + (size_t)r * NPIX + pbase))[c4];
      *(v4f*)&xs[r * LROW + c4 * 4] = v;
    }
    __syncthreads();

    if (tid < CCH) {
      float a = 0.f;
      for (int i = 0; i < CHUNK; ++i) a += xs[tid * LROW + i];
      srow += a;
    }

    // Each wave: 2 k-steps of K=4 pixels; same fragment serves as A and B.
    for (int s2 = 0; s2 < 2; ++s2) {
      const int kc = (wave * 2 + s2) * 4 + lh * 2;
      v2f f[4];
      for (int g = 0; g < 4; ++g)
        f[g] = *(const v2f*)&xs[(g * 16 + l16) * LROW + kc];
      for (int i = 0; i < 4; ++i)
        for (int j = 0; j < 4; ++j)
          acc[i][j] = __builtin_amdgcn_wmma_f32_16x16x4_f32(
              false, f[i], false, f[j], (short)0, acc[i][j], false, false);
    }
    __syncthreads();
  }

  // Barrier-serialized cross-wave reduction into LDS (no float atomics).
  for (int w = 0; w < 8; ++w) {
    if (wave == w) {
      for (int i = 0; i < 4; ++i)
        for (int j = 0; j < 4; ++j)
          for (int r = 0; r < 8; ++r) {
            int row = i * 16 + r + 8 * lh;
            int col = j * 16 + l16;
            float v = acc[i][j][r];
            gs[row * CCH + col] = (w == 0) ? v : (gs[row * CCH + col] + v);
          }
    }
    __syncthreads();
  }

  float* gp = gpart + ((size_t)b * WGPB + wg) * (CCH * CCH);
  for (int i = 0; i < 16; ++i) gp[tid + 256 * i] = gs[tid + 256 * i];
  if (tid < CCH) spart[((size_t)b * WGPB + wg) * CCH + tid] = srow;
}

// ---------------------------------------------------------------------------
// Kernel 2: tiny per-batch pass. Builds S from (G, s, weights), row-softmax,
// then folds everything into M' = wo·A·wv + I and c = wo·(A·bv) + bo.
// One workgroup per batch; pure VALU (negligible runtime).
// ---------------------------------------------------------------------------
__global__ __launch_bounds__(256) void attn_kernel(
    const float* __restrict__ gpart, const float* __restrict__ spart,
    const float* __restrict__ wq, const float* __restrict__ bq,
    const float* __restrict__ wk, const float* __restrict__ bk,
    const float* __restrict__ wv, const float* __restrict__ bv,
    const float* __restrict__ wo, const float* __restrict__ bo,
    float* __restrict__ Mout, float* __restrict__ cvout) {
  __shared__ float Gs[4096];
  __shared__ float Ws[4096];
  __shared__ float Ts[4096];
  __shared__ float sv[64], va[64], vb[64];

  const int b = blockIdx.x, tid = threadIdx.x;

  for (int i = 0; i < 16; ++i) {
    int e = tid + 256 * i;
    float a = 0.f;
    for (int p = 0; p < 8; ++p) a += gpart[((size_t)b * 8 + p) * 4096 + e];
    Gs[e] = a;
  }
  if (tid < 64) {
    float a = 0.f;
    for (int p = 0; p < 8; ++p) a += spart[((size_t)b * 8 + p) * 64 + tid];
    sv[tid] = a;
  }
  for (int i = 0; i < 16; ++i) Ws[tid + 256 * i] = wq[tid + 256 * i];
  __syncthreads();

  // Ts = wq·G ; va = wq·s
  for (int i = 0; i < 16; ++i) {
    int e = tid + 256 * i, q = e >> 6, d = e & 63;
    float a = 0.f;
    for (int c = 0; c < 64; ++c) a += Ws[q * 64 + c] * Gs[c * 64 + d];
    Ts[e] = a;
  }
  if (tid < 64) {
    float a = 0.f;
    for (int c = 0; c < 64; ++c) a += Ws[tid * 64 + c] * sv[c];
    va[tid] = a;
  }
  __syncthreads();
  for (int i = 0; i < 16; ++i) Ws[tid + 256 * i] = wk[tid + 256 * i];
  __syncthreads();
  if (tid < 64) {
    float a = 0.f;
    for (int c = 0; c < 64; ++c) a += Ws[tid * 64 + c] * sv[c];
    vb[tid] = a;
  }
  __syncthreads();

  // S[q,k] = Σ_d Ts[q,d]·wk[k,d] + va[q]·bk[k] + bq[q]·vb[k] + N·bq[q]·bk[k]
  for (int i = 0; i < 16; ++i) {
    int e = tid + 256 * i, q = e >> 6, k = e & 63;
    float a = 0.f;
    for (int d = 0; d < 64; ++d) a += Ts[q * 64 + d] * Ws[k * 64 + d];
    a += va[q] * bk[k] + bq[q] * vb[k] + 16384.f * bq[q] * bk[k];
    Gs[e] = a;  // G dead after Ts; reuse for S / A
  }
  __syncthreads();

  // Row softmax (over k) + va = A·bv
  if (tid < 64) {
    float m = -3.4e38f;
    for (int k = 0; k < 64; ++k) m = fmaxf(m, Gs[tid * 64 + k]);
    float ssum = 0.f;
    for (int k = 0; k < 64; ++k) {
      float e2 = __expf(Gs[tid * 64 + k] - m);
      Gs[tid * 64 + k] = e2;
      ssum += e2;
    }
    float inv = 1.f / ssum;
    float a = 0.f;
    for (int k = 0; k < 64; ++k) {
      Gs[tid * 64 + k] *= inv;
      a += Gs[tid * 64 + k] * bv[k];
    }
    va[tid] = a;
  }
  __syncthreads();
  for (int i = 0; i < 16; ++i) Ws[tid + 256 * i] = wv[tid + 256 * i];
  __syncthreads();

  // Ts = A·wv
  for (int i = 0; i < 16; ++i) {
    int e = tid + 256 * i, q = e >> 6, c = e & 63;
    float a = 0.f;
    for (int k = 0; k < 64; ++k) a += Gs[q * 64 + k] * Ws[k * 64 + c];
    Ts[e] = a;
  }
  __syncthreads();
  for (int i = 0; i < 16; ++i) Ws[tid + 256 * i] = wo[tid + 256 * i];
  __syncthreads();

  // M' = wo·Ts + I  ;  c = wo·va + bo
  for (int i = 0; i < 16; ++i) {
    int e = tid + 256 * i, o = e >> 6, c = e & 63;
    float a = (o == c) ? 1.f : 0.f;
    for (int q = 0; q < 64; ++q) a += Ws[o * 64 + q] * Ts[q * 64 + c];
    Mout[(size_t)b * 4096 + e] = a;
  }
  if (tid < 64) {
    float a = bo[tid];
    for (int q = 0; q < 64; ++q) a += Ws[tid * 64 + q] * va[q];
    cvout[b * 64 + tid] = a;
  }
}

// ---------------------------------------------------------------------------
// Kernel 3: out = M'·X + c  (residual + biases already folded in).
// M' fragments (whole 64x64) preloaded into 128 VGPRs per lane; per 16-pixel
// tile: 16 k-steps x 4 WMMAs; accumulators initialized with the bias column.
// X reads hit L2 (filled by pass 1); output stores are non-temporal so the
// 134 MB write stream doesn't evict the L2-resident X working set.
// ---------------------------------------------------------------------------
__global__ __launch_bounds__(256) void apply_kernel(const float* __restrict__ x,
                                                    const float* __restrict__ Mp,
                                                    const float* __restrict__ cv,
                                                    float* __restrict__ out) {
  const int b    = blockIdx.x >> 5;     // 32 WGs per batch
  const int wg   = blockIdx.x & 31;
  const int tid  = threadIdx.x;
  const int lane = tid & 31;
  const int wave = tid >> 5;
  const int lh   = lane >> 4;
  const int l16  = lane & 15;

  const float* xb = x + (size_t)b * CCH * NPIX;
  float* ob       = out + (size_t)b * CCH * NPIX;
  const float* Mb = Mp + (size_t)b * 4096;
  const float* cb = cv + (size_t)b * 64;

  // Preload all A-fragments of M': af[o0][s] per lane = M'[o0*16+l16][4s+2*lh+{0,1}]
  v2f af[4][16];
  for (int o = 0; o < 4; ++o)
    for (int s = 0; s < 16; ++s)
      af[o][s] = *(const v2f*)&Mb[(o * 16 + l16) * 64 + s * 4 + 2 * lh];

  // Bias column per lane (loop-invariant).
  float cvr[4][8];
  for (int o = 0; o < 4; ++o)
    for (int r = 0; r < 8; ++r) cvr[o][r] = cb[o * 16 + r + 8 * lh];

  // 512 pixels per WG -> 32 tiles of 16; each wave handles 4 tiles.
  for (int t = wave; t < 32; t += 8) {
    const int n0 = wg * 512 + t * 16;
    v8f acc[4];
    for (int o = 0; o < 4; ++o)
      for (int r = 0; r < 8; ++r) acc[o][r] = cvr[o][r];  // bias init

    for (int s = 0; s < 16; ++s) {
      const int c0 = s * 4 + 2 * lh;
      v2f bf;
      bf.x = xb[(size_t)c0 * NPIX + n0 + l16];
      bf.y = xb[(size_t)(c0 + 1) * NPIX + n0 + l16];
      for (int o = 0; o < 4; ++o)
        acc[o] = __builtin_amdgcn_wmma_f32_16x16x4_f32(
            false, af[o][s], false, bf, (short)0, acc[o], false, false);
    }
    for (int o = 0; o < 4; ++o)
      for (int r = 0; r < 8; ++r)
        __builtin_nontemporal_store(
            acc[o][r], &ob[(size_t)(o * 16 + r + 8 * lh) * NPIX + n0 + l16]);
  }
}

// ---------------------------------------------------------------------------
extern "C" void kernel_launch(void* const* d_in, const int* in_sizes, int n_in,
                              void* d_out, int out_size, void* d_ws, size_t ws_size,
                              hipStream_t stream) {
  const float* x  = (const float*)d_in[0];
  const float* wq = (const float*)d_in[1];
  const float* bq = (const float*)d_in[2];
  const float* wk = (const float*)d_in[3];
  const float* bk = (const float*)d_in[4];
  const float* wv = (const float*)d_in[5];
  const float* bv = (const float*)d_in[6];
  const float* wo = (const float*)d_in[7];
  const float* bo = (const float*)d_in[8];
  float* out = (float*)d_out;

  float* ws    = (float*)d_ws;
  float* gpart = ws;                                // 32*8*4096
  float* spart = gpart + (size_t)NB * WGPB * 4096;  // 32*8*64
  float* Mp    = spart + (size_t)NB * WGPB * 64;    // 32*4096
  float* cvec  = Mp + (size_t)NB * 4096;            // 32*64

  gram_kernel<<<dim3(NB * WGPB), dim3(256), 0, stream>>>(x, gpart, spart);
  attn_kernel<<<dim3(NB), dim3(256), 0, stream>>>(gpart, spart, wq, bq, wk, bk,
                                                  wv, bv, wo, bo, Mp, cvec);
  apply_kernel<<<dim3(NB * 32), dim3(256), 0, stream>>>(x, Mp, cvec, out);
}